// Decoder_57604101374359
// MI455X (gfx1250) — compile-verified
//
#include <hip/hip_runtime.h>
#include <hip/hip_bf16.h>

// ---------------------------------------------------------------------------
// CDNA5 (gfx1250) implementation of the Graph-WaveNet decoder.
// Heavy GEMMs (TCN causal convs, GAT node projections) run on
// V_WMMA_F32_16X16X4_F32 (fp32 A/B, fp32 accum) to match the fp32 reference.
// ---------------------------------------------------------------------------

typedef __attribute__((ext_vector_type(2))) float v2f;
typedef __attribute__((ext_vector_type(8))) float v8f;

#define DEV static __device__ __forceinline__

constexpr int Bq     = 8;
constexpr int LATENT = 256;
constexpr int Tn     = 128;
constexpr int Nn     = 768;
constexpr int Fn     = 4;
constexpr int Ln     = 3;
constexpr int BT     = Bq * Tn;   // 1024 rows (b,t)
constexpr int Mn     = Bq * Nn;   // 6144 graph nodes
constexpr float ALPHA = 0.2f;

DEV v8f wmma4(v2f a, v2f b, v8f c) {
  // D = A(16x4 f32) * B(4x16 f32) + C(16x16 f32)
  return __builtin_amdgcn_wmma_f32_16x16x4_f32(
      /*neg_a=*/false, a, /*neg_b=*/false, b,
      /*c_mod=*/(short)0, c, /*reuse_a=*/false, /*reuse_b=*/false);
}

DEV float lrelu(float x)    { return x > 0.f ? x : ALPHA * x; }
DEV float sigmoid_f(float x){ return 1.f / (1.f + expf(-x)); }

// Monotonic float<->uint mapping for atomicMax-based segment max.
DEV unsigned encf(float x) {
  unsigned u = __float_as_uint(x);
  return (u & 0x80000000u) ? ~u : (u | 0x80000000u);
}
DEV float decf(unsigned u) {
  unsigned v = (u & 0x80000000u) ? (u & 0x7FFFFFFFu) : ~u;
  return __uint_as_float(v);
}

// ---------------------------------------------------------------------------
// Shunt: dense -> same conv1d -> same conv2d
// ---------------------------------------------------------------------------
__global__ __launch_bounds__(256)
void shunt_dense(const float* __restrict__ x, const float* __restrict__ w,
                 const float* __restrict__ b, float* __restrict__ y0) {
  int idx = blockIdx.x * 256 + threadIdx.x;            // over BT
  if (idx >= BT) return;
  int bb = idx / Tn, t = idx - bb * Tn;
  float acc = b[t];
  for (int l = 0; l < LATENT; ++l)
    acc += x[bb * LATENT + l] * w[l * Tn + t];
  y0[idx] = acc;
}

__global__ __launch_bounds__(256)
void shunt_conv1(const float* __restrict__ y0, const float* __restrict__ w1,
                 const float* __restrict__ b1, float* __restrict__ y1) {
  int idx = blockIdx.x * 256 + threadIdx.x;            // over BT*N
  if (idx >= BT * Nn) return;
  int bt = idx / Nn, n = idx - bt * Nn;
  int b = bt / Tn, t = bt - b * Tn;
  float acc = b1[n];
  #pragma unroll
  for (int k = 0; k < 3; ++k) {
    int tt = t + k - 1;
    if (tt >= 0 && tt < Tn) acc += y0[b * Tn + tt] * w1[k * Nn + n];
  }
  y1[idx] = acc;
}

__global__ __launch_bounds__(256)
void shunt_conv2(const float* __restrict__ y1, const float* __restrict__ w2,
                 const float* __restrict__ b2, float* __restrict__ y) {
  int idx = blockIdx.x * 256 + threadIdx.x;            // over BT*N
  if (idx >= BT * Nn) return;
  int bt = idx / Nn, n = idx - bt * Nn;
  int b = bt / Tn, t = bt - b * Tn;
  float acc[Fn];
  #pragma unroll
  for (int f = 0; f < Fn; ++f) acc[f] = b2[f];
  #pragma unroll
  for (int dt = 0; dt < 3; ++dt) {
    int tt = t + dt - 1;
    if (tt < 0 || tt >= Tn) continue;
    #pragma unroll
    for (int dn = 0; dn < 3; ++dn) {
      int nn = n + dn - 1;
      if (nn < 0 || nn >= Nn) continue;
      float v = y1[(size_t)(b * Tn + tt) * Nn + nn];
      #pragma unroll
      for (int f = 0; f < Fn; ++f) acc[f] += v * w2[(dt * 3 + dn) * Fn + f];
    }
  }
  float4 o = make_float4(acc[0], acc[1], acc[2], acc[3]);
  *(float4*)(y + (size_t)idx * Fn) = o;
}

// ---------------------------------------------------------------------------
// [B,T,N,F] -> [F,BT,N] split (contiguous per-feature planes for the TCN)
// ---------------------------------------------------------------------------
__global__ __launch_bounds__(256)
void y_to_xs(const float* __restrict__ y, float* __restrict__ xs) {
  int idx = blockIdx.x * 256 + threadIdx.x;            // over BT*N
  if (idx >= BT * Nn) return;
  float4 v = *(const float4*)(y + (size_t)idx * Fn);
  xs[(size_t)0 * BT * Nn + idx] = v.x;
  xs[(size_t)1 * BT * Nn + idx] = v.y;
  xs[(size_t)2 * BT * Nn + idx] = v.z;
  xs[(size_t)3 * BT * Nn + idx] = v.w;
}

// ---------------------------------------------------------------------------
// Gated causal TCN via fp32 WMMA.
//   g[f,bt,co] = sigmoid(conv_a) * tanh(conv_b)
//   conv[bt,co] = sum_{k,ci} x[bt-4+k, ci] * W[k,ci,co]   (zero pad t<0)
// Block: 256 threads (8 waves), tile 32 rows x 64 cols; each wave owns one
// 16x16 C tile per branch. LDS stages X(32x64) and Wa/Wb(64x64).
// ---------------------------------------------------------------------------
__global__ __launch_bounds__(256)
void tcn_gated_wmma(const float* __restrict__ xs,   // [F,BT,N]
                    const float* __restrict__ wa,   // [F,5,N,N]
                    const float* __restrict__ ba,   // [F,N]
                    const float* __restrict__ wb,   // [F,5,N,N]
                    const float* __restrict__ bb,   // [F,N]
                    float* __restrict__ gout) {     // [F,BT,N]
  __shared__ float sX[32][65];
  __shared__ float sWa[64][64];
  __shared__ float sWb[64][64];

  const int f   = blockIdx.z;
  const int bt0 = blockIdx.x * 32;
  const int n0  = blockIdx.y * 64;
  const int tid = threadIdx.x;
  const int lane = tid & 31, wave = tid >> 5;
  const int rowSub = wave & 1;     // which 16-row half
  const int colSub = wave >> 1;    // which 16-col quarter
  const int h = lane >> 4, l15 = lane & 15;

  const float* xf  = xs + (size_t)f * BT * Nn;
  const float* waf = wa + (size_t)f * 5 * Nn * Nn;
  const float* wbf = wb + (size_t)f * 5 * Nn * Nn;

  v8f accA = {}; v8f accB = {};

  for (int k = 0; k < 5; ++k) {
    for (int c0 = 0; c0 < Nn; c0 += 64) {
      __syncthreads();
      // Stage X tile with causal time shift (k-4).
      for (int p = tid; p < 32 * 64; p += 256) {
        int r = p >> 6, c = p & 63;
        int bt = bt0 + r;
        int ts = (bt % Tn) + k - 4;
        sX[r][c] = (ts >= 0) ? xf[(size_t)(bt + k - 4) * Nn + (c0 + c)] : 0.f;
      }
      // Stage weight tiles [64 ci][64 co] for both branches.
      for (int p = tid; p < 64 * 64; p += 256) {
        int r = p >> 6, c = p & 63;
        size_t off = ((size_t)k * Nn + (c0 + r)) * Nn + (n0 + c);
        sWa[r][c] = waf[off];
        sWb[r][c] = wbf[off];
      }
      // Prefetch next weight chunk (global_prefetch_b8).
      {
        int nk = k, nc0 = c0 + 64;
        if (nc0 >= Nn) { nc0 = 0; nk = k + 1; }
        if (nk < 5) {
          size_t noff = ((size_t)nk * Nn + nc0) * Nn + n0 + (tid & 63);
          __builtin_prefetch(&waf[noff], 0, 1);
          __builtin_prefetch(&wbf[noff], 0, 1);
        }
      }
      __syncthreads();
      #pragma unroll
      for (int jc = 0; jc < 16; ++jc) {
        int kk = jc * 4 + 2 * h;                       // K = vgpr + 2*half
        v2f a, fa, fb;
        a.x  = sX[rowSub * 16 + l15][kk];
        a.y  = sX[rowSub * 16 + l15][kk + 1];
        fa.x = sWa[kk][colSub * 16 + l15];
        fa.y = sWa[kk + 1][colSub * 16 + l15];
        fb.x = sWb[kk][colSub * 16 + l15];
        fb.y = sWb[kk + 1][colSub * 16 + l15];
        accA = wmma4(a, fa, accA);
        accB = wmma4(a, fb, accB);
      }
    }
  }

  int col = n0 + colSub * 16 + l15;
  float biasA = ba[f * Nn + col];
  float biasB = bb[f * Nn + col];
  float* go = gout + (size_t)f * BT * Nn;
  #pragma unroll
  for (int r = 0; r < 8; ++r) {                        // C: M = r + 8*half
    int bt = bt0 + rowSub * 16 + r + 8 * h;
    float av = sigmoid_f(accA[r] + biasA);
    float bv = tanhf(accB[r] + biasB);
    go[(size_t)bt * Nn + col] = av * bv;
  }
}

// ---------------------------------------------------------------------------
// g[F,BT,N] -> nodes[F, B*N, T]   (LDS-tiled transpose per (b,f))
// ---------------------------------------------------------------------------
__global__ __launch_bounds__(256)
void g_to_nodes(const float* __restrict__ g, float* __restrict__ nodes) {
  __shared__ float tile[32][33];
  int tb = blockIdx.x;             // T/32
  int nb = blockIdx.y;             // N/32
  int bf = blockIdx.z;             // B*F
  int f = bf / Bq, b = bf - f * Bq;
  int tx = threadIdx.x & 31, ty = threadIdx.x >> 5;    // ty in 0..7
  #pragma unroll
  for (int i = 0; i < 4; ++i) {
    int t = tb * 32 + ty + 8 * i, n = nb * 32 + tx;
    tile[ty + 8 * i][tx] = g[((size_t)f * BT + b * Tn + t) * Nn + n];
  }
  __syncthreads();
  #pragma unroll
  for (int i = 0; i < 4; ++i) {
    int n = nb * 32 + ty + 8 * i, t = tb * 32 + tx;
    nodes[((size_t)f * Mn + b * Nn + n) * Tn + t] = tile[tx][ty + 8 * i];
  }
}

// ---------------------------------------------------------------------------
// GAT projections: Q=lrelu(X*qw+qb), K=lrelu(X*kw+kb), V=X*vw via fp32 WMMA.
// One 16-row strip per block; 8 waves = 8 column tiles of T=128.
// ---------------------------------------------------------------------------
__global__ __launch_bounds__(256)
void gat_qkv_wmma(const float* __restrict__ nodes,  // [F,M,T]
                  const float* __restrict__ qw, const float* __restrict__ qb,
                  const float* __restrict__ kw, const float* __restrict__ kb,
                  const float* __restrict__ vw,
                  float* __restrict__ Q, float* __restrict__ Ko,
                  float* __restrict__ V) {
  __shared__ float sA[16][Tn + 1];
  const int f  = blockIdx.y;
  const int m0 = blockIdx.x * 16;
  const int tid = threadIdx.x, lane = tid & 31, wave = tid >> 5;
  const int h = lane >> 4, l15 = lane & 15;

  const float* nf = nodes + ((size_t)f * Mn + m0) * Tn;
  for (int p = tid; p < 16 * Tn; p += 256)
    sA[p >> 7][p & 127] = nf[(size_t)(p >> 7) * Tn + (p & 127)];
  __syncthreads();

  const float* qwf = qw + (size_t)f * Tn * Tn;
  const float* kwf = kw + (size_t)f * Tn * Tn;
  const float* vwf = vw + (size_t)f * Tn * Tn;
  const int nc = wave * 16 + l15;

  v8f aq = {}, ak = {}, av = {};
  for (int jc = 0; jc < Tn / 4; ++jc) {
    int kk = jc * 4 + 2 * h;
    v2f a, fq, fk, fv;
    a.x = sA[l15][kk]; a.y = sA[l15][kk + 1];
    fq.x = qwf[(size_t)kk * Tn + nc]; fq.y = qwf[(size_t)(kk + 1) * Tn + nc];
    fk.x = kwf[(size_t)kk * Tn + nc]; fk.y = kwf[(size_t)(kk + 1) * Tn + nc];
    fv.x = vwf[(size_t)kk * Tn + nc]; fv.y = vwf[(size_t)(kk + 1) * Tn + nc];
    aq = wmma4(a, fq, aq);
    ak = wmma4(a, fk, ak);
    av = wmma4(a, fv, av);
  }
  float biq = qb[f * Tn + nc], bik = kb[f * Tn + nc];
  #pragma unroll
  for (int r = 0; r < 8; ++r) {
    size_t m = (size_t)m0 + r + 8 * h;
    size_t idx = ((size_t)f * Mn + m) * Tn + nc;
    Q[idx]  = lrelu(aq[r] + biq);
    Ko[idx] = lrelu(ak[r] + bik);
    V[idx]  = av[r];
  }
}

// ---------------------------------------------------------------------------
// Edge softmax (segment max / sum) + scatter aggregation
// ---------------------------------------------------------------------------
__global__ __launch_bounds__(256)
void gat_init(float* __restrict__ H, float* __restrict__ den,
              unsigned* __restrict__ smaxU) {
  size_t idx = (size_t)blockIdx.x * 256 + threadIdx.x;
  size_t tot = (size_t)Fn * Mn * Tn;
  for (size_t i = idx; i < tot; i += (size_t)gridDim.x * 256) {
    H[i] = 0.f;
    if (i < (size_t)Fn * Mn) { den[i] = 0.f; smaxU[i] = 0u; }
  }
}

DEV void edge_rc(const int* edges, int E, int e, int& row, int& col) {
  if (e < E) { row = edges[e]; col = edges[E + e]; }
  else       { row = e - E;    col = e - E; }           // self loops
}

__global__ __launch_bounds__(256)
void gat_score(const float* __restrict__ Q, const float* __restrict__ K,
               const int* __restrict__ edges, int E,
               float* __restrict__ score, unsigned* __restrict__ smaxU) {
  int Ep = E + Mn;
  int idx = blockIdx.x * 256 + threadIdx.x;
  if (idx >= Fn * Ep) return;
  int f = idx / Ep, e = idx - f * Ep;
  int row, col; edge_rc(edges, E, e, row, col);
  const float4* q = (const float4*)(Q + ((size_t)f * Mn + row) * Tn);
  const float4* k = (const float4*)(K + ((size_t)f * Mn + col) * Tn);
  float s = 0.f;
  #pragma unroll 4
  for (int i = 0; i < Tn / 4; ++i) {
    float4 a = q[i], b = k[i];
    s += a.x * b.x + a.y * b.y + a.z * b.z + a.w * b.w;
  }
  s /= sqrtf((float)Tn);
  score[idx] = s;
  atomicMax(&smaxU[f * Mn + row], encf(s));
}

__global__ __launch_bounds__(256)
void gat_exden(const float* __restrict__ score, const unsigned* __restrict__ smaxU,
               const int* __restrict__ edges, int E,
               float* __restrict__ ex, float* __restrict__ den) {
  int Ep = E + Mn;
  int idx = blockIdx.x * 256 + threadIdx.x;
  if (idx >= Fn * Ep) return;
  int f = idx / Ep, e = idx - f * Ep;
  int row, col; edge_rc(edges, E, e, row, col);
  float sm = decf(smaxU[f * Mn + row]);
  float v = expf(score[idx] - sm);
  ex[idx] = v;
  atomicAdd(&den[f * Mn + row], v);
}

__global__ __launch_bounds__(128)
void gat_scatter(const float* __restrict__ V, const float* __restrict__ ex,
                 const float* __restrict__ den, const int* __restrict__ edges,
                 int E, float* __restrict__ H) {
  int Ep = E + Mn;
  int e = blockIdx.x, f = blockIdx.y, t = threadIdx.x;
  int row, col; edge_rc(edges, E, e, row, col);
  float att = ex[(size_t)f * Ep + e] / den[f * Mn + row];
  float v = V[((size_t)f * Mn + col) * Tn + t] * att;
  atomicAdd(&H[((size_t)f * Mn + row) * Tn + t], v);
}

// ---------------------------------------------------------------------------
// y_next = H + vb + 1x1 res conv; skip[l] = lrelu(<y_next, skip_w> + skip_b)
// ---------------------------------------------------------------------------
__global__ __launch_bounds__(256)
void layer_combine(const float* __restrict__ H,     // [F,M,T]
                   const float* __restrict__ vb,    // [F,T]
                   const float* __restrict__ ycur,  // [BT,N,F]
                   const float* __restrict__ rw,    // [F,F]  (in,out)
                   const float* __restrict__ rb,    // [F]
                   const float* __restrict__ skw,   // [F]
                   const float* __restrict__ skb,   // [L]
                   int lidx,
                   float* __restrict__ ynext, float* __restrict__ sbuf) {
  int idx = blockIdx.x * 256 + threadIdx.x;          // over BT*N
  if (idx >= BT * Nn) return;
  int bt = idx / Nn, n = idx - bt * Nn;
  int b = bt / Tn, t = bt - b * Tn;
  float4 v = *(const float4*)(ycur + (size_t)idx * Fn);
  float yc[4] = {v.x, v.y, v.z, v.w};
  float yn[4];
  #pragma unroll
  for (int f = 0; f < Fn; ++f) {
    float r = rb[f];
    #pragma unroll
    for (int g = 0; g < Fn; ++g) r += yc[g] * rw[g * Fn + f];
    float hv = H[((size_t)f * Mn + b * Nn + n) * Tn + t];
    yn[f] = hv + vb[f * Tn + t] + r;
  }
  *(float4*)(ynext + (size_t)idx * Fn) = make_float4(yn[0], yn[1], yn[2], yn[3]);
  float sk = skb[lidx];
  #pragma unroll
  for (int f = 0; f < Fn; ++f) sk += yn[f] * skw[f];
  sbuf[(size_t)idx * Ln + lidx] = lrelu(sk);
}

// ---------------------------------------------------------------------------
// Output head: 3x3 SAME conv over (t,n) on skips [B,T,N,L] -> lrelu -> 1x1
// ---------------------------------------------------------------------------
__global__ __launch_bounds__(256)
void out_head(const float* __restrict__ sbuf, const float* __restrict__ w1,
              const float* __restrict__ b1, const float* __restrict__ w2,
              const float* __restrict__ b2, float* __restrict__ out) {
  int idx = blockIdx.x * 256 + threadIdx.x;          // over BT*N
  if (idx >= BT * Nn) return;
  int bt = idx / Nn, n = idx - bt * Nn;
  int b = bt / Tn, t = bt - b * Tn;
  float acc = b1[0];
  #pragma unroll
  for (int dt = 0; dt < 3; ++dt) {
    int tt = t + dt - 1;
    if (tt < 0 || tt >= Tn) continue;
    #pragma unroll
    for (int dn = 0; dn < 3; ++dn) {
      int nn = n + dn - 1;
      if (nn < 0 || nn >= Nn) continue;
      const float* sp = sbuf + ((size_t)(b * Tn + tt) * Nn + nn) * Ln;
      const float* wp = w1 + (dt * 3 + dn) * Ln;
      acc += sp[0] * wp[0] + sp[1] * wp[1] + sp[2] * wp[2];
    }
  }
  out[idx] = lrelu(acc) * w2[0] + b2[0];
}

// ---------------------------------------------------------------------------
// Host orchestration (all launches on `stream`; workspace ~98 MB)
// ---------------------------------------------------------------------------
extern "C" void kernel_launch(void* const* d_in, const int* in_sizes, int n_in,
                              void* d_out, int out_size, void* d_ws, size_t ws_size,
                              hipStream_t stream) {
  const float* x        = (const float*)d_in[0];
  const int*   edges    = (const int*)  d_in[1];
  const float* sd_w     = (const float*)d_in[2];
  const float* sd_b     = (const float*)d_in[3];
  const float* sc1_w    = (const float*)d_in[4];
  const float* sc1_b    = (const float*)d_in[5];
  const float* sc2_w    = (const float*)d_in[6];
  const float* sc2_b    = (const float*)d_in[7];
  const float* tcn_a_w  = (const float*)d_in[8];
  const float* tcn_a_b  = (const float*)d_in[9];
  const float* tcn_b_w  = (const float*)d_in[10];
  const float* tcn_b_b  = (const float*)d_in[11];
  const float* gat_q_w  = (const float*)d_in[12];
  const float* gat_q_b  = (const float*)d_in[13];
  const float* gat_k_w  = (const float*)d_in[14];
  const float* gat_k_b  = (const float*)d_in[15];
  const float* gat_v_w  = (const float*)d_in[16];
  const float* gat_v_b  = (const float*)d_in[17];
  const float* res_w    = (const float*)d_in[18];
  const float* res_b    = (const float*)d_in[19];
  const float* skip_w   = (const float*)d_in[20];
  const float* skip_b   = (const float*)d_in[21];
  const float* out1_w   = (const float*)d_in[22];
  const float* out1_b   = (const float*)d_in[23];
  const float* out2_w   = (const float*)d_in[24];
  const float* out2_b   = (const float*)d_in[25];
  (void)n_in; (void)out_size; (void)ws_size;

  const int E  = in_sizes[1] / 2;      // 12288
  const int Ep = E + Mn;               // + self loops

  float* ws = (float*)d_ws;
  const size_t SZ = (size_t)BT * Nn * Fn;   // 3,145,728 floats
  float* ycur   = ws;
  float* ynext  = ws + SZ;
  float* xs     = ws + 2 * SZ;   // also reused as nodes [F,M,T]
  float* gbuf   = ws + 3 * SZ;   // also reused as H     [F,M,T]
  float* Qb     = ws + 4 * SZ;
  float* Kb     = ws + 5 * SZ;
  float* Vb     = ws + 6 * SZ;
  float* sbuf   = ws + 7 * SZ;                       // [BT,N,L]
  float* scoreb = sbuf + (size_t)BT * Nn * Ln;
  float* exb    = scoreb + (size_t)Fn * Ep;
  float* denb   = exb + (size_t)Fn * Ep;
  unsigned* smaxU = (unsigned*)(denb + (size_t)Fn * Mn);
  float* y0     = (float*)(smaxU + (size_t)Fn * Mn); // [BT]
  float* y1     = y0 + BT;                           // [BT,N]

  const int btn_blocks = (BT * Nn + 255) / 256;

  shunt_dense<<<(BT + 255) / 256, 256, 0, stream>>>(x, sd_w, sd_b, y0);
  shunt_conv1<<<btn_blocks, 256, 0, stream>>>(y0, sc1_w, sc1_b, y1);
  shunt_conv2<<<btn_blocks, 256, 0, stream>>>(y1, sc2_w, sc2_b, ycur);

  for (int l = 0; l < Ln; ++l) {
    y_to_xs<<<btn_blocks, 256, 0, stream>>>(ycur, xs);

    dim3 gt(BT / 32, Nn / 64, Fn);
    tcn_gated_wmma<<<gt, 256, 0, stream>>>(
        xs,
        tcn_a_w + (size_t)l * Fn * 5 * Nn * Nn, tcn_a_b + (size_t)l * Fn * Nn,
        tcn_b_w + (size_t)l * Fn * 5 * Nn * Nn, tcn_b_b + (size_t)l * Fn * Nn,
        gbuf);

    dim3 gn(Tn / 32, Nn / 32, Bq * Fn);
    g_to_nodes<<<gn, 256, 0, stream>>>(gbuf, xs /* nodes */);

    dim3 gq(Mn / 16, Fn);
    gat_qkv_wmma<<<gq, 256, 0, stream>>>(
        xs,
        gat_q_w + (size_t)l * Fn * Tn * Tn, gat_q_b + (size_t)l * Fn * Tn,
        gat_k_w + (size_t)l * Fn * Tn * Tn, gat_k_b + (size_t)l * Fn * Tn,
        gat_v_w + (size_t)l * Fn * Tn * Tn,
        Qb, Kb, Vb);

    gat_init<<<1024, 256, 0, stream>>>(gbuf /* H */, denb, smaxU);

    int etot = Fn * Ep;
    gat_score<<<(etot + 255) / 256, 256, 0, stream>>>(Qb, Kb, edges, E, scoreb, smaxU);
    gat_exden<<<(etot + 255) / 256, 256, 0, stream>>>(scoreb, smaxU, edges, E, exb, denb);
    dim3 gs(Ep, Fn);
    gat_scatter<<<gs, Tn, 0, stream>>>(Vb, exb, denb, edges, E, gbuf);

    layer_combine<<<btn_blocks, 256, 0, stream>>>(
        gbuf, gat_v_b + (size_t)l * Fn * Tn, ycur,
        res_w + (size_t)l * Fn * Fn, res_b + (size_t)l * Fn,
        skip_w + (size_t)l * Fn, skip_b, l, ynext, sbuf);

    float* tmp = ycur; ycur = ynext; ynext = tmp;
  }

  out_head<<<btn_blocks, 256, 0, stream>>>(sbuf, out1_w, out1_b, out2_w, out2_b,
                                           (float*)d_out);
}